// RoIPooling_58918361366608
// MI455X (gfx1250) — compile-verified
//
#include <hip/hip_runtime.h>

typedef float v2f __attribute__((ext_vector_type(2)));
typedef float v8f __attribute__((ext_vector_type(8)));

#define RPI    256   // ROIs per image
#define CCH    1024  // channels
#define FH     50
#define FW     50
#define NWAVES 8

// S(14x14) = Y(14x16) * P(16x16) * X^T(16x14), padded to 16, per (roi, channel).
// Stage 1: T = P * X^T   (A = patch, B = x-weights)
// Stage 2: S = Y * T     (A = y-weights, B = T via LDS relayout)
// Then 2x2 max-pool -> 7x7, packed through LDS for coalesced stores.
__launch_bounds__(256)
__global__ void roialign_wmma_kernel(const float* __restrict__ rois,
                                     const float* __restrict__ fm,
                                     float* __restrict__ out)
{
    __shared__ float lds_patch[NWAVES][256];
    __shared__ float lds_t[NWAVES][256];

    const int tid  = threadIdx.x;
    const int wave = tid >> 5;
    const int l    = tid & 31;
    const int hf   = l >> 4;   // lane half (0: lanes 0-15, 1: lanes 16-31)
    const int lm   = l & 15;

    const int bi = blockIdx.x;
    const int n  = bi / RPI;
    const int r  = bi % RPI;

    const float* roi = rois + (n * RPI + r) * 4;
    const float y1 = roi[0] * 0.0625f;
    const float x1 = roi[1] * 0.0625f;
    const float y2 = (roi[0] + roi[2]) * 0.0625f;
    const float x2 = (roi[1] + roi[3]) * 0.0625f;
    const float ystep = (y2 - y1) * (1.0f / 13.0f);
    const float xstep = (x2 - x1) * (1.0f / 13.0f);
    const int y_start = (int)floorf(y1);
    const int x_start = (int)floorf(x1);

    // Per-lane WMMA weight operands, chunks k=0..3 (K = 4k + 2*hf + {0,1}).
    // xw[k] = stage-1 B = X^T rows;  yw[k] = stage-2 A = Y rows.
    v2f xw[4], yw[4];
    {
        const float px  = x1 + (float)lm * xstep;
        const float fx0 = floorf(px);
        const float frx = px - fx0;
        const int   icx = (int)fx0 - x_start;
        const float py  = y1 + (float)lm * ystep;
        const float fy0 = floorf(py);
        const float fry = py - fy0;
        const int   icy = (int)fy0 - y_start;
        const float vmask = (lm < 14) ? 1.0f : 0.0f;  // pad rows 14,15 to zero
#pragma unroll
        for (int k = 0; k < 4; ++k) {
            const int c0 = 4 * k + 2 * hf;
            const int c1 = c0 + 1;
            float wx0 = (c0 == icx) ? (1.0f - frx) : ((c0 == icx + 1) ? frx : 0.0f);
            float wx1 = (c1 == icx) ? (1.0f - frx) : ((c1 == icx + 1) ? frx : 0.0f);
            float wy0 = (c0 == icy) ? (1.0f - fry) : ((c0 == icy + 1) ? fry : 0.0f);
            float wy1 = (c1 == icy) ? (1.0f - fry) : ((c1 == icy + 1) ? fry : 0.0f);
            xw[k].x = wx0 * vmask; xw[k].y = wx1 * vmask;
            yw[k].x = wy0 * vmask; yw[k].y = wy1 * vmask;
        }
    }

    float* patch = lds_patch[wave];
    float* tbuf  = lds_t[wave];

    const long fm_img = (long)n * CCH * (FH * FW);

    for (int c = wave; c < CCH; c += NWAVES) {
        const float* fmc = fm + fm_img + (long)c * (FH * FW);

        // ---- fill 16x16 zero-padded patch (coalesced 64B row reads) ----
#pragma unroll
        for (int it = 0; it < 8; ++it) {
            const int e   = it * 32 + l;
            const int row = e >> 4, col = e & 15;
            const int y = y_start + row, x = x_start + col;
            const int yc = min(y, FH - 1), xc = min(x, FW - 1);
            float v = fmc[yc * FW + xc];
            v = (y <= FH - 1 && x <= FW - 1) ? v : 0.0f;  // zero-pad OOB
            patch[e] = v;
        }
        __builtin_amdgcn_wave_barrier();

        // ---- stage 1: T = P x X^T  (four chained 16x16x4 f32 WMMAs) ----
        v8f t1 = {0.f, 0.f, 0.f, 0.f, 0.f, 0.f, 0.f, 0.f};
#pragma unroll
        for (int k = 0; k < 4; ++k) {
            v2f a = *(const v2f*)&patch[lm * 16 + 4 * k + 2 * hf];  // A: P[lm][K..K+1]
            t1 = __builtin_amdgcn_wmma_f32_16x16x4_f32(
                false, a, false, xw[k], (short)0, t1, false, false);
        }
        // D-layout -> LDS as T[rrow][j]
#pragma unroll
        for (int vv = 0; vv < 8; ++vv)
            tbuf[(vv + 8 * hf) * 16 + lm] = t1[vv];
        __builtin_amdgcn_wave_barrier();

        // ---- stage 2: S = Y x T ----
        v8f s = {0.f, 0.f, 0.f, 0.f, 0.f, 0.f, 0.f, 0.f};
#pragma unroll
        for (int k = 0; k < 4; ++k) {
            v2f b;
            b.x = tbuf[(4 * k + 2 * hf) * 16 + lm];      // B: T[K][lm]
            b.y = tbuf[(4 * k + 2 * hf + 1) * 16 + lm];  // B: T[K+1][lm]
            s = __builtin_amdgcn_wmma_f32_16x16x4_f32(
                false, yw[k], false, b, (short)0, s, false, false);
        }

        // ---- 2x2 max-pool, stage through LDS ----
        // lane holds S[8*hf + 0..7][lm]; row-pair max -> tbuf[t*32 + l]
#pragma unroll
        for (int t = 0; t < 4; ++t)
            tbuf[t * 32 + l] = fmaxf(s[2 * t], s[2 * t + 1]);
        __builtin_amdgcn_wave_barrier();

        // ---- 49 outputs, two coalesced dword stores ----
        float* ob = out + ((long)((n * RPI + r) * CCH + c)) * 49;
        {
            const int e  = l;                    // 0..31
            const int oy = e / 7, ox = e % 7;
            const int idx = (oy < 4) ? (oy * 32 + 2 * ox)
                                     : ((oy - 4) * 32 + 16 + 2 * ox);
            ob[e] = fmaxf(tbuf[idx], tbuf[idx + 1]);
        }
        if (l < 17) {
            const int e  = l + 32;               // 32..48
            const int oy = e / 7, ox = e % 7;
            const int idx = (oy < 4) ? (oy * 32 + 2 * ox)
                                     : ((oy - 4) * 32 + 16 + 2 * ox);
            ob[e] = fmaxf(tbuf[idx], tbuf[idx + 1]);
        }
        __builtin_amdgcn_wave_barrier();
    }
}

extern "C" void kernel_launch(void* const* d_in, const int* in_sizes, int n_in,
                              void* d_out, int out_size, void* d_ws, size_t ws_size,
                              hipStream_t stream) {
    (void)in_sizes; (void)n_in; (void)out_size; (void)d_ws; (void)ws_size;
    const float* rois = (const float*)d_in[0];       // [4,256,4]
    const float* fm   = (const float*)d_in[1];       // [4,1024,50,50]
    float* out        = (float*)d_out;               // [4,256,1024,7,7]
    const int nblocks = 4 * RPI;                     // one block per ROI
    roialign_wmma_kernel<<<nblocks, 256, 0, stream>>>(rois, fm, out);
}